// QMWLNPairwiseAtomClassifier_54391465837030
// MI455X (gfx1250) — compile-verified
//
#include <hip/hip_runtime.h>
#include <hip/hip_bf16.h>

// Problem constants (match reference)
#define B_     16
#define NR_    96
#define NG_    64
#define AF_    98
#define BFW_   6
#define H_     300
#define QM_    160
#define MAXNB_ 10
#define P_     2048
#define F_     760      // 2*H + QM
#define D0_    770      // F + 10
#define PPB_   128      // P / B rows per segment (contiguous)

typedef float v2f __attribute__((ext_vector_type(2)));
typedef float v8f __attribute__((ext_vector_type(8)));

enum { GF_RELU = 1, GF_ACC = 2, GF_MULC = 4 };

// ---------------------------------------------------------------------------
// fp32 WMMA GEMM: C(MxN) = op( A(MxK) @ B(KxN) [+C] ), B given TRANSPOSED as
// Bt (N x K, row-major) so both A and B fragments are contiguous v2f loads.
// One wave per block computes a 64x64 output strip (16 v8f accumulators).
// Per K-step: 4x global_load_b64 (A) + 4x global_load_b64 (B) -> 16 WMMA.
// K-loop is software-pipelined (double-buffered fragments). M/N tails are
// handled by CLAMPING load addresses (WMMA has no cross-row / cross-col
// mixing; clamped rows/cols are skipped by the guarded store), so the hot
// loop is branch- and select-free. Only the K%4 tail is one guarded step.
// flags: GF_ACC  -> init accumulator from C (split-K concat GEMMs)
//        GF_RELU -> relu at store
//        GF_MULC -> multiply result by existing C at store (h_nei fusion)
// ---------------------------------------------------------------------------
#define WMMA1(rt, t, av, bv)                                                   \
    acc[rt][t] = __builtin_amdgcn_wmma_f32_16x16x4_f32(                        \
        false, av, false, bv, (short)0, acc[rt][t], false, false)

#define WMMA_ALL(a0_, a1_, a2_, a3_, b0_, b1_, b2_, b3_)                       \
    do {                                                                       \
        WMMA1(0, 0, a0_, b0_); WMMA1(1, 0, a1_, b0_);                          \
        WMMA1(2, 0, a2_, b0_); WMMA1(3, 0, a3_, b0_);                          \
        WMMA1(0, 1, a0_, b1_); WMMA1(1, 1, a1_, b1_);                          \
        WMMA1(2, 1, a2_, b1_); WMMA1(3, 1, a3_, b1_);                          \
        WMMA1(0, 2, a0_, b2_); WMMA1(1, 2, a1_, b2_);                          \
        WMMA1(2, 2, a2_, b2_); WMMA1(3, 2, a3_, b2_);                          \
        WMMA1(0, 3, a0_, b3_); WMMA1(1, 3, a1_, b3_);                          \
        WMMA1(2, 3, a2_, b3_); WMMA1(3, 3, a3_, b3_);                          \
    } while (0)

__global__ __launch_bounds__(32)
void gemm_wmma_f32(const float* __restrict__ A, int lda,
                   const float* __restrict__ Bt, int ldbt,
                   float* __restrict__ C, int ldc,
                   int M, int K, int N, int flags)
{
    const int lane = threadIdx.x;
    const int row0 = blockIdx.x * 64;
    const int col0 = blockIdx.y * 64;
    const int half = lane >> 4;     // 0: K offsets {0,1}, 1: {2,3}
    const int l16  = lane & 15;

    // clamped row / col indices (clamped rows/cols are never stored)
    int rowAc[4], colc[4];
#pragma unroll
    for (int i = 0; i < 4; ++i) {
        const int r = row0 + 16 * i + l16;
        rowAc[i] = r < M ? r : (M - 1);
        const int c = col0 + 16 * i + l16;
        colc[i] = c < N ? c : (N - 1);
    }

    v8f acc[4][4];
    if (flags & GF_ACC) {
#pragma unroll
        for (int rt = 0; rt < 4; ++rt)
#pragma unroll
            for (int t = 0; t < 4; ++t)
#pragma unroll
                for (int r = 0; r < 8; ++r) {
                    const int row  = row0 + 16 * rt + r + 8 * half;
                    const int rowc = row < M ? row : (M - 1);
                    acc[rt][t][r] = C[(size_t)rowc * ldc + colc[t]];
                }
    } else {
#pragma unroll
        for (int rt = 0; rt < 4; ++rt)
#pragma unroll
            for (int t = 0; t < 4; ++t) acc[rt][t] = (v8f)0.f;
    }

    // named scalar pointers (arrays defeat global-addrspace inference)
    const float* ap0 = A + (size_t)rowAc[0] * lda + 2 * half;
    const float* ap1 = A + (size_t)rowAc[1] * lda + 2 * half;
    const float* ap2 = A + (size_t)rowAc[2] * lda + 2 * half;
    const float* ap3 = A + (size_t)rowAc[3] * lda + 2 * half;
    const float* bp0 = Bt + (size_t)colc[0] * ldbt + 2 * half;
    const float* bp1 = Bt + (size_t)colc[1] * ldbt + 2 * half;
    const float* bp2 = Bt + (size_t)colc[2] * ldbt + 2 * half;
    const float* bp3 = Bt + (size_t)colc[3] * ldbt + 2 * half;

    const int K4 = K & ~3;

    v2f a0, a1, a2, a3, b0, b1, b2, b3;
    if (K4 > 0) {
        a0 = *(const v2f*)ap0; ap0 += 4;
        a1 = *(const v2f*)ap1; ap1 += 4;
        a2 = *(const v2f*)ap2; ap2 += 4;
        a3 = *(const v2f*)ap3; ap3 += 4;
        b0 = *(const v2f*)bp0; bp0 += 4;
        b1 = *(const v2f*)bp1; bp1 += 4;
        b2 = *(const v2f*)bp2; bp2 += 4;
        b3 = *(const v2f*)bp3; bp3 += 4;
    }
#pragma unroll 2
    for (int k0 = 4; k0 < K4; k0 += 4) {
        // issue next step's loads first (latency hidden behind 16 WMMAs)
        v2f A0 = *(const v2f*)ap0; ap0 += 4;
        v2f A1 = *(const v2f*)ap1; ap1 += 4;
        v2f A2 = *(const v2f*)ap2; ap2 += 4;
        v2f A3 = *(const v2f*)ap3; ap3 += 4;
        v2f B0 = *(const v2f*)bp0; bp0 += 4;
        v2f B1 = *(const v2f*)bp1; bp1 += 4;
        v2f B2 = *(const v2f*)bp2; bp2 += 4;
        v2f B3 = *(const v2f*)bp3; bp3 += 4;
        WMMA_ALL(a0, a1, a2, a3, b0, b1, b2, b3);
        a0 = A0; a1 = A1; a2 = A2; a3 = A3;
        b0 = B0; b1 = B1; b2 = B2; b3 = B3;
    }
    if (K4 > 0) {
        WMMA_ALL(a0, a1, a2, a3, b0, b1, b2, b3);
    }

    // K tail: single guarded step (zero the out-of-range K slice)
    if (K4 < K) {
        const int  ka  = K4 + 2 * half;
        const int  k0c = ka       < K ? ka       : (K - 1);
        const int  k1c = (ka + 1) < K ? (ka + 1) : (K - 1);
        const bool v0  = ka < K, v1 = (ka + 1) < K;
        v2f at[4], bt[4];
#pragma unroll
        for (int i = 0; i < 4; ++i) {
            const float* ab = A + (size_t)rowAc[i] * lda;
            at[i].x = v0 ? ab[k0c] : 0.f;
            at[i].y = v1 ? ab[k1c] : 0.f;
            const float* bb = Bt + (size_t)colc[i] * ldbt;
            bt[i].x = v0 ? bb[k0c] : 0.f;
            bt[i].y = v1 ? bb[k1c] : 0.f;
        }
#pragma unroll
        for (int t = 0; t < 4; ++t)
#pragma unroll
            for (int rt = 0; rt < 4; ++rt)
                acc[rt][t] = __builtin_amdgcn_wmma_f32_16x16x4_f32(
                    false, at[rt], false, bt[t], (short)0, acc[rt][t], false, false);
    }

    // guarded store + epilogue
#pragma unroll
    for (int rt = 0; rt < 4; ++rt)
#pragma unroll
        for (int t = 0; t < 4; ++t) {
            const int col = col0 + 16 * t + l16;
            if (col >= N) continue;
#pragma unroll
            for (int r = 0; r < 8; ++r) {
                const int row = row0 + 16 * rt + r + 8 * half;
                if (row >= M) continue;
                float v = acc[rt][t][r];
                if (flags & GF_RELU) v = fmaxf(v, 0.f);
                if (flags & GF_MULC) v *= C[(size_t)row * ldc + col];
                C[(size_t)row * ldc + col] = v;
            }
        }
}

// ------------------------ weight transpose (one-shot) -----------------------
// W: K x N (row-major)  ->  WT: N x K (row-major)
__global__ void transpose_w(const float* __restrict__ W, float* __restrict__ WT,
                            int K, int N)
{
    const int t = blockIdx.x * blockDim.x + threadIdx.x;
    const int total = K * N;
    if (t >= total) return;
    const int k = t / N, n = t - k * N;
    WT[(size_t)n * K + k] = W[t];
}

// ------------------------ elementwise / gather helpers ----------------------
__global__ void gather_fa(const float* __restrict__ af, const int* __restrict__ graph,
                          float* __restrict__ fa, int Nn, int nrows)
{
    const int t = blockIdx.x * blockDim.x + threadIdx.x;
    const int total = nrows * H_;
    if (t >= total) return;
    const int row = t / H_, h = t - row * H_;
    const int gb = graph[row * 2 + 0];
    const int gn = graph[row * 2 + 1];
    fa[(size_t)row * H_ + h] = af[((size_t)gb * Nn + gn) * H_ + h];
}

__global__ void gather_fb(const float* __restrict__ bond, const int* __restrict__ graph,
                          float* __restrict__ fb, int Nn, int nrows)
{
    const int t = blockIdx.x * blockDim.x + threadIdx.x;
    const int total = nrows * BFW_;
    if (t >= total) return;
    const int row = t / BFW_, c = t - row * BFW_;
    const int gb = graph[row * 2 + 0];
    const int gn = graph[row * 2 + 1];
    fb[(size_t)row * BFW_ + c] = bond[((size_t)gb * Nn + gn) * BFW_ + c];
}

// out[rn,h] = sum_{nb < num_nbs[rn]} in[rn,nb,h]
__global__ void masked_sum(const float* __restrict__ hin, const int* __restrict__ num_nbs,
                           float* __restrict__ out, int rows)
{
    const int t = blockIdx.x * blockDim.x + threadIdx.x;
    const int total = rows * H_;
    if (t >= total) return;
    const int rn = t / H_, h = t - rn * H_;
    int nn = num_nbs[rn];
    if (nn > MAXNB_) nn = MAXNB_;
    float s = 0.f;
    for (int nb = 0; nb < nn; ++nb)
        s += hin[((size_t)rn * MAXNB_ + nb) * H_ + h];
    out[(size_t)rn * H_ + h] = s;
}

// kernels = f_nei * t_self * node_mask
__global__ void mul_mask(const float* __restrict__ a, const float* __restrict__ b,
                         const float* __restrict__ mask, float* __restrict__ out, int rows)
{
    const int t = blockIdx.x * blockDim.x + threadIdx.x;
    const int total = rows * H_;
    if (t >= total) return;
    const int rn = t / H_;
    out[t] = a[t] * b[t] * mask[rn];
}

// pair[(i*NG+j), h] = res_h[b,i,h] + rg_h[b,j,h]   (one batch b)
__global__ void pair_add(const float* __restrict__ res_h, const float* __restrict__ rg_h,
                         float* __restrict__ pairb, int b)
{
    const int t = blockIdx.x * blockDim.x + threadIdx.x;
    const int total = NR_ * NG_ * H_;
    if (t >= total) return;
    const int row = t / H_, h = t - row * H_;
    const int i = row / NG_, j = row - i * NG_;
    pairb[t] = res_h[((size_t)(b * NR_ + i)) * H_ + h] +
               rg_h [((size_t)(b * NG_ + j)) * H_ + h];
}

// att[row] = sigmoid( t[row,:] . Ws )   — one wave per row
__global__ void att_sigmoid(const float* __restrict__ tatt, const float* __restrict__ Ws,
                            float* __restrict__ att_out, int rows)
{
    const int wid  = (blockIdx.x * blockDim.x + threadIdx.x) >> 5;
    const int lane = threadIdx.x & 31;
    if (wid >= rows) return;
    float s = 0.f;
    for (int k = lane; k < H_; k += 32) s += tatt[(size_t)wid * H_ + k] * Ws[k];
#pragma unroll
    for (int off = 16; off; off >>= 1) s += __shfl_xor(s, off, 32);
    if (lane == 0) att_out[wid] = 1.f / (1.f + __expf(-s));
}

// ctx[b,i,h] = sum_j att[b,i,j] * rg_h[b,j,h]
__global__ void ctx_kernel(const float* __restrict__ att, const float* __restrict__ rg_h,
                           float* __restrict__ ctx)
{
    const int t = blockIdx.x * blockDim.x + threadIdx.x;
    const int total = B_ * NR_ * H_;
    if (t >= total) return;
    const int bi = t / H_, h = t - bi * H_;
    const int b = bi / NR_;
    float s = 0.f;
    for (int j = 0; j < NG_; ++j)
        s += att[(size_t)bi * NG_ + j] * rg_h[((size_t)(b * NG_ + j)) * H_ + h];
    ctx[t] = s;
}

// h760 = concat([res_h, ctx, fatom_qm], -1)
__global__ void concat_h(const float* __restrict__ res_h, const float* __restrict__ ctxv,
                         const float* __restrict__ fq, float* __restrict__ h760)
{
    const int t = blockIdx.x * blockDim.x + threadIdx.x;
    const int total = B_ * NR_ * F_;
    if (t >= total) return;
    const int bi = t / F_, d = t - bi * F_;
    float v;
    if (d < H_)            v = res_h[(size_t)bi * H_ + d];
    else if (d < 2 * H_)   v = ctxv [(size_t)bi * H_ + (d - H_)];
    else                   v = fq   [(size_t)bi * QM_ + (d - 2 * H_)];
    h760[t] = v;
}

// As[p,:] = concat( h[seg,i]+h[seg,j], connect[p] )
__global__ void build_score_rows(const float* __restrict__ h760, const int* __restrict__ cm,
                                 const float* __restrict__ connect, float* __restrict__ As)
{
    const int t = blockIdx.x * blockDim.x + threadIdx.x;
    const int total = P_ * D0_;
    if (t >= total) return;
    const int p = t / D0_, d = t - p * D0_;
    const int seg = cm[p * 3 + 0];
    const int i   = cm[p * 3 + 1];
    const int j   = cm[p * 3 + 2];
    float v;
    if (d < F_)
        v = h760[((size_t)(seg * NR_ + i)) * F_ + d] +
            h760[((size_t)(seg * NR_ + j)) * F_ + d];
    else
        v = connect[(size_t)p * 10 + (d - F_)];
    As[t] = v;
}

// mean[b,d] = (1/128) * sum_{p in segment b} rh[p,d]  (contiguous segments)
__global__ void seg_mean(const float* __restrict__ rh, float* __restrict__ meanb)
{
    const int t = blockIdx.x * blockDim.x + threadIdx.x;
    const int total = B_ * D0_;
    if (t >= total) return;
    const int b = t / D0_, d = t - b * D0_;
    float s = 0.f;
    for (int p = 0; p < PPB_; ++p)
        s += rh[((size_t)(b * PPB_ + p)) * D0_ + d];
    meanb[t] = s * (1.f / (float)PPB_);
}

// out[b] = mean[b,:] . W_score   — one wave per batch
__global__ __launch_bounds__(32)
void final_gemv(const float* __restrict__ meanb, const float* __restrict__ Ws,
                float* __restrict__ out)
{
    const int b = blockIdx.x;
    const int lane = threadIdx.x;
    float s = 0.f;
    for (int d = lane; d < D0_; d += 32) s += meanb[(size_t)b * D0_ + d] * Ws[d];
#pragma unroll
    for (int off = 16; off; off >>= 1) s += __shfl_xor(s, off, 32);
    if (lane == 0) out[b] = s;
}

// ---------------------------------------------------------------------------
extern "C" void kernel_launch(void* const* d_in, const int* in_sizes, int n_in,
                              void* d_out, int out_size, void* d_ws, size_t ws_size,
                              hipStream_t stream)
{
    (void)in_sizes; (void)n_in; (void)out_size; (void)ws_size;

    const float* res_atom   = (const float*)d_in[0];
    const float* res_bond   = (const float*)d_in[1];
    const int*   res_ag     = (const int*)  d_in[2];
    const int*   res_bg     = (const int*)  d_in[3];
    const int*   res_nnb    = (const int*)  d_in[4];
    const float* res_nmask  = (const float*)d_in[5];
    const float* rg_atom    = (const float*)d_in[6];
    const float* rg_bond    = (const float*)d_in[7];
    const int*   rg_ag      = (const int*)  d_in[8];
    const int*   rg_bg      = (const int*)  d_in[9];
    const int*   rg_nnb     = (const int*)  d_in[10];
    const float* rg_nmask   = (const float*)d_in[11];
    const int*   core_mask  = (const int*)  d_in[12];
    const float* fatom_qm   = (const float*)d_in[13];
    const float* connect    = (const float*)d_in[14];
    const float* res_W_atom = (const float*)d_in[15];
    const float* res_W_na   = (const float*)d_in[16];
    const float* res_W_nb   = (const float*)d_in[17];
    const float* res_W_self = (const float*)d_in[18];
    const float* res_W_U2   = (const float*)d_in[19];
    const float* res_W_U1   = (const float*)d_in[20];
    const float* rg_W_atom  = (const float*)d_in[21];
    const float* rg_W_na    = (const float*)d_in[22];
    const float* rg_W_nb    = (const float*)d_in[23];
    const float* rg_W_self  = (const float*)d_in[24];
    const float* rg_W_U2    = (const float*)d_in[25];
    const float* rg_W_U1    = (const float*)d_in[26];
    const float* W_att_h    = (const float*)d_in[27];
    const float* W_att_s    = (const float*)d_in[28];
    const float* W_score0   = (const float*)d_in[29];
    const float* W_score    = (const float*)d_in[30];

    // ---- workspace carve (bump allocator, 256B aligned) ----
    char* wsb = (char*)d_ws;
    size_t off = 0;
    auto carve = [&](size_t nElems) -> float* {
        float* p = (float*)(wsb + off);
        off += ((nElems * sizeof(float)) + 255) & ~(size_t)255;
        return p;
    };
    float* kern_res = carve((size_t)B_ * NR_ * H_);        // res_h (WLN output)
    float* kern_rg  = carve((size_t)B_ * NG_ * H_);        // rg_h
    float* ctxv     = carve((size_t)B_ * NR_ * H_);
    float* h760     = carve((size_t)B_ * NR_ * F_);
    float* As       = carve((size_t)P_ * D0_);
    float* rh       = carve((size_t)P_ * D0_);
    float* meanb    = carve((size_t)B_ * D0_);
    float* attf     = carve((size_t)B_ * NR_ * NG_);
    float* afA      = carve((size_t)B_ * NR_ * H_);
    float* afB      = carve((size_t)B_ * NR_ * H_);
    float* fa       = carve((size_t)B_ * NR_ * MAXNB_ * H_);
    float* fb       = carve((size_t)B_ * NR_ * MAXNB_ * BFW_);
    float* pre      = carve((size_t)B_ * NR_ * MAXNB_ * H_);
    float* nlab     = carve((size_t)B_ * NR_ * H_);
    float* fnei     = carve((size_t)B_ * NR_ * H_);
    float* tself    = carve((size_t)B_ * NR_ * H_);
    float* pairb    = carve((size_t)NR_ * NG_ * H_);
    float* tatt     = carve((size_t)NR_ * NG_ * H_);
    // transposed weights (N x K)
    float* WaT_res    = carve((size_t)H_ * AF_);
    float* WnaT_res   = carve((size_t)H_ * H_);
    float* WnbT_res   = carve((size_t)H_ * BFW_);
    float* WselfT_res = carve((size_t)H_ * H_);
    float* WU2T_res   = carve((size_t)H_ * (H_ + BFW_));
    float* WU1T_res   = carve((size_t)H_ * (2 * H_));
    float* WaT_rg     = carve((size_t)H_ * AF_);
    float* WnaT_rg    = carve((size_t)H_ * H_);
    float* WnbT_rg    = carve((size_t)H_ * BFW_);
    float* WselfT_rg  = carve((size_t)H_ * H_);
    float* WU2T_rg    = carve((size_t)H_ * (H_ + BFW_));
    float* WU1T_rg    = carve((size_t)H_ * (2 * H_));
    float* WattT      = carve((size_t)H_ * H_);
    float* Ws0T       = carve((size_t)D0_ * D0_);

    auto launch1d = [&](int total) { return dim3((total + 255) / 256); };
    auto tr = [&](const float* W, float* WT, int K, int N) {
        transpose_w<<<launch1d(K * N), 256, 0, stream>>>(W, WT, K, N);
    };

    // one-shot weight transposes
    tr(res_W_atom, WaT_res,    AF_,       H_);
    tr(res_W_na,   WnaT_res,   H_,        H_);
    tr(res_W_nb,   WnbT_res,   BFW_,      H_);
    tr(res_W_self, WselfT_res, H_,        H_);
    tr(res_W_U2,   WU2T_res,   H_ + BFW_, H_);
    tr(res_W_U1,   WU1T_res,   2 * H_,    H_);
    tr(rg_W_atom,  WaT_rg,     AF_,       H_);
    tr(rg_W_na,    WnaT_rg,    H_,        H_);
    tr(rg_W_nb,    WnbT_rg,    BFW_,      H_);
    tr(rg_W_self,  WselfT_rg,  H_,        H_);
    tr(rg_W_U2,    WU2T_rg,    H_ + BFW_, H_);
    tr(rg_W_U1,    WU1T_rg,    2 * H_,    H_);
    tr(W_att_h,    WattT,      H_,        H_);
    tr(W_score0,   Ws0T,       D0_,       D0_);

    auto gemm = [&](const float* A, int lda, const float* Bt, int ldbt,
                    float* C, int ldc, int M, int K, int N, int flags) {
        dim3 grid((M + 63) / 64, (N + 63) / 64);
        gemm_wmma_f32<<<grid, dim3(32), 0, stream>>>(A, lda, Bt, ldbt, C, ldc, M, K, N, flags);
    };

    // ---- WLN (res / rg). Only the last depth iteration needs the kernels
    // path (W_nei_*, W_self); earlier iterations only need the af update. ----
    auto run_wln = [&](const float* atom, const float* bond,
                       const int* ag, const int* bg, const int* nnb,
                       const float* nmask,
                       const float* WaT, const float* WnaT, const float* WnbT,
                       const float* WselfT, const float* WU2T, const float* WU1T,
                       int Nn, float* kern_out) {
        const int rows  = B_ * Nn;
        const int nrows = rows * MAXNB_;
        // af = relu(atom @ W_atom)
        gemm(atom, AF_, WaT, AF_, afA, H_, rows, AF_, H_, GF_RELU);
        float* cur = afA;
        float* nxt = afB;
        for (int it = 0; it < 4; ++it) {
            gather_fa<<<launch1d(nrows * H_), 256, 0, stream>>>(cur, ag, fa, Nn, nrows);
            gather_fb<<<launch1d(nrows * BFW_), 256, 0, stream>>>(bond, bg, fb, Nn, nrows);
            if (it < 3) {
                // pre = relu([fa|fb] @ W_U2)  (split-K accumulate)
                gemm(fa, H_,   WU2T,      H_ + BFW_, pre, H_, nrows, H_,   H_, 0);
                gemm(fb, BFW_, WU2T + H_, H_ + BFW_, pre, H_, nrows, BFW_, H_, GF_ACC | GF_RELU);
                masked_sum<<<launch1d(rows * H_), 256, 0, stream>>>(pre, nnb, nlab, rows);
                // af' = relu([af|nei_label] @ W_U1)
                gemm(cur,  H_, WU1T,      2 * H_, nxt, H_, rows, H_, H_, 0);
                gemm(nlab, H_, WU1T + H_, 2 * H_, nxt, H_, rows, H_, H_, GF_ACC | GF_RELU);
                float* tmp = cur; cur = nxt; nxt = tmp;
            } else {
                // h_nei = relu(fa@Wna) * relu(fb@Wnb)  (MULC fusion into pre)
                gemm(fa, H_,   WnaT, H_,   pre, H_, nrows, H_,   H_, GF_RELU);
                gemm(fb, BFW_, WnbT, BFW_, pre, H_, nrows, BFW_, H_, GF_RELU | GF_MULC);
                masked_sum<<<launch1d(rows * H_), 256, 0, stream>>>(pre, nnb, fnei, rows);
                gemm(cur, H_, WselfT, H_, tself, H_, rows, H_, H_, GF_RELU);
                mul_mask<<<launch1d(rows * H_), 256, 0, stream>>>(fnei, tself, nmask, kern_out, rows);
            }
        }
    };

    run_wln(res_atom, res_bond, res_ag, res_bg, res_nnb, res_nmask,
            WaT_res, WnaT_res, WnbT_res, WselfT_res, WU2T_res, WU1T_res,
            NR_, kern_res);
    run_wln(rg_atom, rg_bond, rg_ag, rg_bg, rg_nnb, rg_nmask,
            WaT_rg, WnaT_rg, WnbT_rg, WselfT_rg, WU2T_rg, WU1T_rg,
            NG_, kern_rg);

    // ---- attention, chunked per batch to bound workspace ----
    const int arows = NR_ * NG_;
    for (int b = 0; b < B_; ++b) {
        pair_add<<<launch1d(arows * H_), 256, 0, stream>>>(kern_res, kern_rg, pairb, b);
        gemm(pairb, H_, WattT, H_, tatt, H_, arows, H_, H_, GF_RELU);
        att_sigmoid<<<launch1d(arows * 32), 256, 0, stream>>>(tatt, W_att_s, attf + (size_t)b * arows, arows);
    }
    ctx_kernel<<<launch1d(B_ * NR_ * H_), 256, 0, stream>>>(attf, kern_rg, ctxv);
    concat_h<<<launch1d(B_ * NR_ * F_), 256, 0, stream>>>(kern_res, ctxv, fatom_qm, h760);

    // ---- scoring head ----
    build_score_rows<<<launch1d(P_ * D0_), 256, 0, stream>>>(h760, core_mask, connect, As);
    gemm(As, D0_, Ws0T, D0_, rh, D0_, P_, D0_, D0_, GF_RELU);
    seg_mean<<<launch1d(B_ * D0_), 256, 0, stream>>>(rh, meanb);
    final_gemv<<<dim3(B_), dim3(32), 0, stream>>>(meanb, W_score, (float*)d_out);
}